// Node_func_55155970015731
// MI455X (gfx1250) — compile-verified
//
#include <hip/hip_runtime.h>

// Fused ragged gather-sum-add:
//   out[i, :] = sub[i, :] + x[idx[i,0], :] + x[idx[i,1], :]
// S = 500000, D = 128, fp32. Pure streaming kernel, HBM-bound (~0.56 GB/run
// after x becomes L2-resident -> ~24 us floor at 23.3 TB/s).
//
// Strategy (MI455X / gfx1250, wave32):
//  - one wave per row: 32 lanes x float4 = 512 B = one full 128-float row
//    -> coalesced global_load_b128 / global_store_b128
//  - NT temporal hints on the single-use streams (sub read, out write) so the
//    51.2 MB x table stays resident in the 192 MB L2 for the random gather
//  - x gather loads use default (RT) policy so they cache in L2

typedef float v4f __attribute__((ext_vector_type(4)));
typedef int   v2i __attribute__((ext_vector_type(2)));

#ifndef D_DIM
#define D_DIM 128
#endif

#define ROWS_PER_BLOCK 8   // 256 threads / 32-lane waves

__global__ __launch_bounds__(256) void gather_sum_add_kernel(
    const float* __restrict__ x,      // [N_NODES, 128]
    const float* __restrict__ sub,    // [S, 128]
    const int*   __restrict__ idx,    // [S, 2] (int32 per harness contract)
    float*       __restrict__ out,    // [S, 128]
    int n_subs)
{
    const int lane   = threadIdx.x & 31;
    const int waveId = threadIdx.x >> 5;
    const long long row = (long long)blockIdx.x * ROWS_PER_BLOCK + waveId;
    if (row >= n_subs) return;

    // Both gather indices in one 8-byte load (global_load_b64).
    const v2i p = *(const v2i*)(idx + row * 2);

    const v4f* __restrict__ srow = (const v4f*)(sub + row * (long long)D_DIM);
    const v4f* __restrict__ xr0  = (const v4f*)(x + (long long)p.x * D_DIM);
    const v4f* __restrict__ xr1  = (const v4f*)(x + (long long)p.y * D_DIM);
    v4f*       __restrict__ orow = (v4f*)(out + row * (long long)D_DIM);

    // Streamed-once operand: non-temporal load (th:TH_LOAD_NT) -> don't
    // pollute L2, keep it for the x gather table.
    v4f s = __builtin_nontemporal_load(srow + lane);
    // Gathered rows: regular (RT) loads so x stays L2-resident.
    v4f a = xr0[lane];
    v4f b = xr1[lane];

    v4f r = s + a + b;

    // Streamed-once result: non-temporal store (th:TH_STORE_NT).
    __builtin_nontemporal_store(r, orow + lane);
}

extern "C" void kernel_launch(void* const* d_in, const int* in_sizes, int n_in,
                              void* d_out, int out_size, void* d_ws, size_t ws_size,
                              hipStream_t stream) {
    const float* x   = (const float*)d_in[0];   // [100000,128] fp32
    const float* sub = (const float*)d_in[1];   // [500000,128] fp32
    const int*   idx = (const int*)  d_in[2];   // [500000,2] int32
    float*       out = (float*)d_out;           // [500000,128] fp32

    const int n_subs = in_sizes[1] / D_DIM;     // 500000
    const int blocks = (n_subs + ROWS_PER_BLOCK - 1) / ROWS_PER_BLOCK;

    gather_sum_add_kernel<<<blocks, 256, 0, stream>>>(x, sub, idx, out, n_subs);
}